// ProductWithExternalWeights_20813411516869
// MI455X (gfx1250) — compile-verified
//
#include <hip/hip_runtime.h>
#include <stdint.h>

// ProductWithExternalWeights on MI455X (gfx1250):
// out[b,u] = prod_k ( x[b,k]*W[k,u] + 1 - W[k,u] ) = prod_k fma(W[k,u], x[b,k]-1, 1)
// VALU-bound (~0.8 GFLOP vs ~0.3us of HBM traffic at 23.3 TB/s).
// Strategy: minimum VALU ops/elem using VOP3P packed f32 (v_pk_fma_f32 + v_pk_mul_f32
// = 1 instruction per 2 elements each), weights staged via CDNA5 async-to-LDS DMA
// (ASYNCcnt) double-buffered and overlapped with compute, x-1 precomputed in LDS.

typedef float v2f __attribute__((ext_vector_type(2)));

namespace {
constexpr int kB     = 2048;
constexpr int kInDim = 256;
constexpr int kUnits = 512;

constexpr int TILE_B  = 16;            // rows per block: 2 per wave x 8 waves
constexpr int TILE_U  = 128;           // units per block: 4 per lane x 32 lanes
constexpr int KC      = 32;            // k-chunk per LDS buffer
constexpr int NCHUNK  = kInDim / KC;   // 8
constexpr int THREADS = 256;           // 8 wave32
}

// ---- CDNA5 async global->LDS (tracked by ASYNCcnt, CDNA5 ISA) ----
__device__ __forceinline__ void async_ld_b128(uint32_t lds_byte_addr, const float* gaddr) {
    // GV mode: vdst = LDS byte address VGPR, vaddr = 64-bit global address VGPRs
    asm volatile("global_load_async_to_lds_b128 %0, %1, off"
                 :: "v"(lds_byte_addr), "v"(gaddr)
                 : "memory");
}
__device__ __forceinline__ void wait_async0() {
    asm volatile("s_wait_asynccnt 0" ::: "memory");
}

__global__ __launch_bounds__(THREADS)
void prod_ext_w_kernel(const float* __restrict__ x,
                       const float* __restrict__ w,
                       float* __restrict__ out)
{
    __shared__ __align__(16) float shW[2][KC * TILE_U];     // 2 x 16 KB weight chunks
    __shared__ __align__(16) float shX[TILE_B * kInDim];    // 16 KB x tile (holds x-1)

    const int tid  = threadIdx.x;
    const int lane = tid & 31;
    const int wid  = tid >> 5;

    const int u0 = blockIdx.x * TILE_U;
    const int b0 = blockIdx.y * TILE_B;

    // ---- stage full x tile (16x256 f32 = 1024 float4, 4 per thread) via async DMA ----
    #pragma unroll
    for (int it = 0; it < 4; ++it) {
        const int idx4 = tid + it * THREADS;     // 0..1023
        const int r    = idx4 >> 6;              // 64 float4 per row
        const int c4   = idx4 & 63;
        const float* g = x + (size_t)(b0 + r) * kInDim + c4 * 4;
        const uint32_t l = (uint32_t)(size_t)&shX[r * kInDim + c4 * 4];
        async_ld_b128(l, g);
    }

    // ---- weight-chunk stager: KC x TILE_U = 1024 float4, 4 per thread ----
    auto stage_w = [&](int buf, int chunk) {
        const int k0 = chunk * KC;
        #pragma unroll
        for (int it = 0; it < 4; ++it) {
            const int idx4 = tid + it * THREADS;
            const int kl   = idx4 >> 5;          // 32 float4 per k-row (128 floats)
            const int u4   = idx4 & 31;
            const float* g = w + (size_t)(k0 + kl) * kUnits + u0 + u4 * 4;
            const uint32_t l = (uint32_t)(size_t)&shW[buf][kl * TILE_U + u4 * 4];
            async_ld_b128(l, g);
        }
    };

    stage_w(0, 0);
    wait_async0();          // this wave's x tile + chunk 0 landed in LDS
    __syncthreads();        // everyone's data visible

    // ---- one-time in-place transform: shX <- x - 1 (hoists subtract out of hot loop) ----
    {
        float4* p = (float4*)shX;
        #pragma unroll
        for (int it = 0; it < 4; ++it) {
            const int i = tid + it * THREADS;    // 0..1023 float4s
            float4 v = p[i];
            v.x -= 1.0f; v.y -= 1.0f; v.z -= 1.0f; v.w -= 1.0f;
            p[i] = v;
        }
    }
    __syncthreads();

    const int r0 = 2 * wid;
    const int r1 = r0 + 1;

    const v2f one = {1.0f, 1.0f};
    v2f acc00 = one, acc01 = one;   // row r0, units [lane*4 .. +3]
    v2f acc10 = one, acc11 = one;   // row r1

    for (int c = 0; c < NCHUNK; ++c) {
        if (c + 1 < NCHUNK) stage_w((c + 1) & 1, c + 1);   // overlap DMA with compute

        const float* Wb = shW[c & 1];
        const float* X0 = &shX[r0 * kInDim + c * KC];      // already x-1
        const float* X1 = &shX[r1 * kInDim + c * KC];

        #pragma unroll 4
        for (int kl = 0; kl < KC; ++kl) {
            const float xs0 = X0[kl];                      // wave-uniform LDS broadcast
            const float xs1 = X1[kl];
            const v2f xm0 = {xs0, xs0};
            const v2f xm1 = {xs1, xs1};

            const float4 wv = *(const float4*)&Wb[kl * TILE_U + lane * 4];
            const v2f wa = {wv.x, wv.y};
            const v2f wb = {wv.z, wv.w};

            // contracts to v_pk_fma_f32; products to v_pk_mul_f32 (VOP3P)
            acc00 *= wa * xm0 + one;
            acc01 *= wb * xm0 + one;
            acc10 *= wa * xm1 + one;
            acc11 *= wb * xm1 + one;
        }

        wait_async0();      // next chunk resident (no-op on final iteration)
        __syncthreads();    // all waves done reading current buffer before reuse
    }

    float4 o0 = make_float4(acc00.x, acc00.y, acc01.x, acc01.y);
    float4 o1 = make_float4(acc10.x, acc10.y, acc11.x, acc11.y);
    *(float4*)(out + (size_t)(b0 + r0) * kUnits + u0 + lane * 4) = o0;
    *(float4*)(out + (size_t)(b0 + r1) * kUnits + u0 + lane * 4) = o1;
}

extern "C" void kernel_launch(void* const* d_in, const int* in_sizes, int n_in,
                              void* d_out, int out_size, void* d_ws, size_t ws_size,
                              hipStream_t stream) {
    const float* x   = (const float*)d_in[0];   // [2048, 256] f32
    const float* wts = (const float*)d_in[1];   // [256*512]   f32 (row-major [k, u])
    float* out       = (float*)d_out;           // [2048, 512] f32

    dim3 grid(kUnits / TILE_U, kB / TILE_B);    // (4, 128) = 512 blocks
    prod_ext_w_kernel<<<grid, THREADS, 0, stream>>>(x, wts, out);
}